// DifferentiableExtrusion_6622839570469
// MI455X (gfx1250) — compile-verified
//
#include <hip/hip_runtime.h>
#include <math.h>

#define VV 64
#define NB 8
#define NP 32
#define PP 16
#define SHARP 100.0f
#define EPSV 1e-8f

typedef float v2f __attribute__((ext_vector_type(2)));
typedef float v8f __attribute__((ext_vector_type(8)));

__global__ __launch_bounds__(128) void extrusion_kernel(
    const float* __restrict__ poly,   // (8,32,16,2)
    const float* __restrict__ attr,   // (8,6)
    const float* __restrict__ valsc,  // (8,32)
    float* __restrict__ out)          // (8,64,64,64)
{
    __shared__ float sRaw[NP * PP * 2];          // 1024 floats, raw vertices
    __shared__ float sX0[NP * PP], sY0[NP * PP];
    __shared__ float sEX[NP * PP], sEY[NP * PP];
    __shared__ float sIQ[NP * PP];               // 1/(|e|^2+eps)
    __shared__ float sIY[NP * PP];               // 1/(ey+eps)
    __shared__ int   sOk[NP];
    __shared__ float sComb[2 * VV];              // 2 rows of combined mask

    const int t    = threadIdx.x;
    const int b    = blockIdx.x >> 5;            // 32 blocks per batch
    const int yblk = (blockIdx.x & 31) << 1;     // 2 rows per block

    // CDNA5: global_prefetch_b8 of this batch's polygon slab
    __builtin_prefetch(poly + b * (NP * PP * 2), 0, 0);

    // ---- Stage raw polygon vertices into LDS (256 float4 / 128 threads) ----
    {
        const float4* s4 = (const float4*)(poly + b * (NP * PP * 2));
        float4* d4 = (float4*)sRaw;
        d4[t]       = s4[t];
        d4[t + 128] = s4[t + 128];
    }
    __syncthreads();

    // ---- Per-polygon validity ----
    if (t < NP) {
        int cnt = 0;
        #pragma unroll
        for (int p = 0; p < PP; ++p) {
            float s = sRaw[(t * PP + p) * 2 + 0] + sRaw[(t * PP + p) * 2 + 1];
            cnt += (s != 0.0f) ? 1 : 0;
        }
        sOk[t] = (valsc[b * NP + t] >= 0.5f) && (cnt >= 3) ? 1 : 0;
    }
    // ---- Per-edge constants (512 edges, 4 per thread) ----
    #pragma unroll
    for (int i = 0; i < 4; ++i) {
        int e = t * 4 + i;
        int n = e >> 4, p = e & 15, p1 = (p + 1) & 15;
        float x0 = sRaw[(n * PP + p ) * 2 + 0];
        float y0 = sRaw[(n * PP + p ) * 2 + 1];
        float x1 = sRaw[(n * PP + p1) * 2 + 0];
        float y1 = sRaw[(n * PP + p1) * 2 + 1];
        float ex = x1 - x0, ey = y1 - y0;
        sX0[e] = x0; sY0[e] = y0; sEX[e] = ex; sEY[e] = ey;
        sIQ[e] = 1.0f / (ex * ex + ey * ey + EPSV);
        sIY[e] = 1.0f / (ey + EPSV);
    }
    __syncthreads();

    // ---- Per-point SDF -> sigmoid -> max over valid polygons ----
    const float inv = 1.0f / (float)(VV - 1);
    const int   y   = yblk + (t >> 6);
    const int   x   = t & 63;
    const float px  = (float)x * inv;
    const float py  = (float)y * inv;

    float combined = 0.0f;
    for (int n = 0; n < NP; ++n) {
        if (!sOk[n]) continue;                   // wave-uniform skip
        float mind2 = 3.402823466e38f;
        int par = 0;
        #pragma unroll
        for (int p = 0; p < PP; ++p) {
            int e = n * PP + p;
            float x0 = sX0[e], y0 = sY0[e], ex = sEX[e], ey = sEY[e];
            float vx = px - x0, vy = py - y0;
            float ve = vx * ex + vy * ey;
            float tt = fminf(fmaxf(ve * sIQ[e], 0.0f), 1.0f);
            float dx = vx - tt * ex, dy = vy - tt * ey;
            float d2 = dx * dx + dy * dy;
            mind2 = fminf(mind2, d2);
            float y1 = y0 + ey;
            bool cross = ((y0 <= py) && (y1 > py)) || ((y1 <= py) && (y0 > py));
            float ix = x0 + ex * ((py - y0) * sIY[e]);
            par ^= (cross && (ix > px)) ? 1 : 0;
        }
        float sdf = sqrtf(mind2);
        sdf = par ? -sdf : sdf;
        float m = 1.0f / (1.0f + __expf(SHARP * sdf));
        combined = fmaxf(combined, m);
    }
    sComb[t] = combined;                         // t == ylocal*64 + x
    __syncthreads();

    // ---- Depth threshold ----
    int h = (int)floorf(attr[b * 6 + 0] * (float)VV);
    h = h < 1 ? 1 : (h > VV ? VV : h);

    // ---- Extrusion: out[d,x] = dmask[d] (x) combined[x], via WMMA f32 16x16x4 ----
    const int wave = t >> 5;
    const int lane = t & 31;
    const int lx   = lane & 15;
    const int hi   = lane >> 4;
    const float hiz = (hi == 0) ? 1.0f : 0.0f;   // zero K>=1 halves branchlessly

    float* outb = out + (size_t)b * (VV * VV * VV);

    #pragma unroll
    for (int i = 0; i < 8; ++i) {
        int tile = wave * 8 + i;                 // 32 tiles: 2y * 4d * 4x
        int yl = tile >> 4;
        int d0 = ((tile >> 2) & 3) * 16;
        int x0 = (tile & 3) * 16;
        int yy = yblk + yl;

        v8f C = {};
#if defined(__gfx1250__) && __has_builtin(__builtin_amdgcn_wmma_f32_16x16x4_f32)
        // A (16x4): K=0 column lives in lanes 0-15, VGPR0 (ISA 7.12.2); K>0 zero.
        // Unconditional LDS read (lx<16 is safe for all lanes) + cndmask, so no
        // divergent EXEC save/restore around the WMMA.
        float dval = ((d0 + lx) < h) ? 1.0f : 0.0f;
        float cval = sComb[yl * VV + x0 + lx];
        v2f Am; Am.x = dval * hiz; Am.y = 0.0f;
        v2f Bm; Bm.x = cval * hiz; Bm.y = 0.0f;
        C = __builtin_amdgcn_wmma_f32_16x16x4_f32(
                false, Am, false, Bm, (short)0, C, false, false);
#else
        float cb = sComb[yl * VV + x0 + lx];
        #pragma unroll
        for (int r = 0; r < 8; ++r)
            C[r] = ((d0 + r + hi * 8) < h ? 1.0f : 0.0f) * cb;
#endif
        // D layout: VGPR r -> row M = r + 8*hi, col N = lx
        int dbase = d0 + hi * 8;
        float* p0 = outb + ((size_t)dbase * VV + yy) * VV + x0 + lx;
        #pragma unroll
        for (int r = 0; r < 8; ++r)
            p0[(size_t)r * (VV * VV)] = C[r];
    }
}

extern "C" void kernel_launch(void* const* d_in, const int* in_sizes, int n_in,
                              void* d_out, int out_size, void* d_ws, size_t ws_size,
                              hipStream_t stream) {
    (void)in_sizes; (void)n_in; (void)out_size; (void)d_ws; (void)ws_size;
    const float* poly  = (const float*)d_in[0];
    const float* attr  = (const float*)d_in[1];
    const float* valsc = (const float*)d_in[2];
    float* out = (float*)d_out;
    // 8 batches * 32 row-pairs = 256 blocks, 128 threads (4 wave32s)
    extrusion_kernel<<<256, 128, 0, stream>>>(poly, attr, valsc, out);
}